// HEGN_Loss_44160853738072
// MI455X (gfx1250) — compile-verified
//
#include <hip/hip_runtime.h>

#pragma float_control(precise, off)

#define BATCH 4
#define NPTS 8192
#define MPTS 8192
#define MSPLIT 2
#define MHALF (MPTS / MSPLIT)                              // 4096 cols per block
#define WAVES_PER_BLOCK 8
#define ROWS_PER_WAVE 32                                   // two 16-row A tiles
#define ROWS_PER_BLOCK (WAVES_PER_BLOCK * ROWS_PER_WAVE)   // 256
#define MTILE 16
#define UBIG 0x7F7FFFFFu   // bit pattern of FLT_MAX; uint order == float order for d >= 0

typedef __attribute__((ext_vector_type(2))) float v2f;
typedef __attribute__((ext_vector_type(8))) float v8f;
typedef __attribute__((ext_vector_type(4))) unsigned int v4u;
typedef __attribute__((ext_vector_type(8))) int v8i;
typedef __attribute__((ext_vector_type(4))) int v4i;

__device__ __forceinline__ unsigned umin32(unsigned a, unsigned b) {
    return a < b ? a : b;   // v_min_u32 / v_min3_u32 — no FP canonicalization
}

__global__ void hegn_init_kernel(unsigned* minbufs) {
    int i = blockIdx.x * blockDim.x + threadIdx.x;
    if (i < BATCH * (MPTS + NPTS)) minbufs[i] = UBIG;
}

__global__ __launch_bounds__(256) void hegn_chamfer_kernel(
    const float* __restrict__ x, const float* __restrict__ y,
    unsigned* __restrict__ colmin, unsigned* __restrict__ rowmin) {
    __shared__ float ylds[MHALF * 3];   // 48KB: y half staged by TDM
    __shared__ unsigned scol[MHALF];    // 16KB: per-block column mins (uint order)

    const int tid = threadIdx.x;
    const int lane = tid & 31;
    const int wave = tid >> 5;
    const int mhalf = blockIdx.x & (MSPLIT - 1);
    const int q = blockIdx.x / MSPLIT;
    const int b = q / (NPTS / ROWS_PER_BLOCK);
    const int chunk = q % (NPTS / ROWS_PER_BLOCK);
    const int mbase = mhalf * MHALF;

    for (int i = tid; i < MHALF; i += 256) scol[i] = UBIG;

    // ---- TDM: stage this block's y range (MHALF points = 48KB) into LDS ----
    const float* ysrc = y + ((size_t)b * MPTS + mbase) * 3;
    if (wave == 0) {
        const unsigned ELEMS = MHALF * 3;  // 12288 dword elements, 1-row tile
        unsigned long long ga = (unsigned long long)(size_t)ysrc;
        unsigned ldsoff = (unsigned)(size_t)&ylds[0];
        v4u g0;
        g0.x = 1u;                                   // count=1 valid descriptor
        g0.y = ldsoff;                               // lds_addr
        g0.z = (unsigned)(ga & 0xFFFFFFFFu);         // global_addr[31:0]
        g0.w = (unsigned)((ga >> 32) & 0x01FFFFFFu)  // global_addr[56:32]
               | 0x80000000u;                        // type=2 (image)
        v8i g1;
        g1[0] = 0x00020000;                          // data_size=2 (4B), mask=0
        g1[1] = (int)((ELEMS & 0xFFFFu) << 16);      // tensor_dim0[15:0]
        g1[2] = (int)(((ELEMS >> 16) & 0xFFFFu) | (1u << 16)); // dim0 hi | tensor_dim1=1
        g1[3] = (int)((ELEMS & 0xFFFFu) << 16);      // tile_dim0 = ELEMS
        g1[4] = 1;                                   // tile_dim1 = 1
        g1[5] = (int)ELEMS;                          // tensor_dim0_stride[31:0]
        g1[6] = 0;
        g1[7] = 0;
        v4i g2 = {0, 0, 0, 0};
        v4i g3 = {0, 0, 0, 0};
        v8i g4 = {0, 0, 0, 0, 0, 0, 0, 0};
        __builtin_amdgcn_tensor_load_to_lds(g0, g1, g2, g3, g4, 0);
        __builtin_amdgcn_s_wait_tensorcnt((short)0);
    }
    __syncthreads();

    // ---- Two augmented A tiles (32 rows of x) + C = ||x||^2 broadcast ----
    // A_row = (-2x0, -2x1, -2x2, 1); C[row][col] = ||x_row||^2.
    // With B_col = (y0, y1, y2, ||y||^2):
    //   D = A*B + C = ||x||^2 + ||y||^2 - 2 x.y  (full squared distance in WMMA)
    // 32-bit A 16x4 layout: lanes 0-15 hold {K0,K1}, lanes 16-31 hold {K2,K3}
    const int row0 = chunk * ROWS_PER_BLOCK + wave * ROWS_PER_WAVE + (lane & 15);
    const int rofs = (lane & 16) >> 1;  // C VGPR i -> row i or i+8
    v2f a[2];
    v8f xc[2];
    unsigned rmn[2][8];
#pragma unroll
    for (int t = 0; t < 2; ++t) {
        const float* xp = x + ((size_t)b * NPTS + row0 + t * 16) * 3;
        float x0 = xp[0], x1 = xp[1], x2 = xp[2];
        float xn = x0 * x0 + x1 * x1 + x2 * x2;
        a[t].x = (lane < 16) ? -2.0f * x0 : -2.0f * x2;
        a[t].y = (lane < 16) ? -2.0f * x1 : 1.0f;
#pragma unroll
        for (int i = 0; i < 8; ++i) {
            xc[t][i] = __shfl(xn, i + rofs, 32);  // ||x_row||^2 in C layout
            rmn[t][i] = UBIG;
        }
    }

    // ---- Main loop: 2 WMMAs per staged 16-column tile; D is the distance ----
    for (int mt = 0; mt < MHALF / MTILE; ++mt) {
        const int lc = mt * MTILE + (lane & 15);
        float y0 = ylds[lc * 3 + 0];
        float y1 = ylds[lc * 3 + 1];
        float y2 = ylds[lc * 3 + 2];
        float yn = y0 * y0 + y1 * y1 + y2 * y2;
        v2f bm;  // B_col = (y0, y1, y2, yn)
        bm.x = (lane < 16) ? y0 : y2;
        bm.y = (lane < 16) ? y1 : yn;

        unsigned cmin = UBIG;
#pragma unroll
        for (int t = 0; t < 2; ++t) {
            v8f c = __builtin_amdgcn_wmma_f32_16x16x4_f32(
                false, a[t], false, bm, (short)0, xc[t], false, false);
#pragma unroll
            for (int i = 0; i < 8; ++i) {
                unsigned du = __float_as_uint(c[i]);  // order-preserving for d>=0
                rmn[t][i] = umin32(rmn[t][i], du);    // running row min
                cmin = umin32(cmin, du);              // col min over 16 rows
            }
        }
        // All 32 lanes min into scol[lc]: lanes l and l+16 cover rows 0-7/8-15
        // of each tile; compiler-tracked ds_min_u32 merges the collision.
        atomicMin(&scol[lc], cmin);
    }

    // ---- Row mins: reduce over columns, publish via global atomic min ----
    unsigned* rowb = rowmin + (size_t)b * NPTS;
#pragma unroll
    for (int t = 0; t < 2; ++t) {
#pragma unroll
        for (int i = 0; i < 8; ++i) {
            unsigned v = rmn[t][i];
            v = umin32(v, (unsigned)__shfl_xor((int)v, 1, 32));
            v = umin32(v, (unsigned)__shfl_xor((int)v, 2, 32));
            v = umin32(v, (unsigned)__shfl_xor((int)v, 4, 32));
            v = umin32(v, (unsigned)__shfl_xor((int)v, 8, 32));
            rmn[t][i] = v;  // lanes 0-15: row t*16+i ; lanes 16-31: row t*16+8+i
        }
        const int rt = chunk * ROWS_PER_BLOCK + wave * ROWS_PER_WAVE + t * 16;
        if (lane == 0) {
#pragma unroll
            for (int i = 0; i < 8; ++i) atomicMin(&rowb[rt + i], rmn[t][i]);
        }
        if (lane == 16) {
#pragma unroll
            for (int i = 0; i < 8; ++i) atomicMin(&rowb[rt + 8 + i], rmn[t][i]);
        }
    }

    // ---- Merge block column mins into global ----
    __syncthreads();
    unsigned* colb = colmin + (size_t)b * MPTS + mbase;
    for (int i = tid; i < MHALF; i += 256) atomicMin(&colb[i], scol[i]);
}

__global__ __launch_bounds__(256) void hegn_finalize_kernel(
    const unsigned* __restrict__ colmin, const unsigned* __restrict__ rowmin,
    const float* __restrict__ R, const float* __restrict__ S,
    const float* __restrict__ t, const float* __restrict__ R_gt,
    const float* __restrict__ S_gt, const float* __restrict__ t_gt,
    float* __restrict__ out) {
    __shared__ float redc[256];
    __shared__ float redr[256];
    float sc = 0.0f, sr = 0.0f;
    for (int i = threadIdx.x; i < BATCH * MPTS; i += 256)
        sc += __uint_as_float(colmin[i]);
    for (int i = threadIdx.x; i < BATCH * NPTS; i += 256)
        sr += __uint_as_float(rowmin[i]);
    redc[threadIdx.x] = sc;
    redr[threadIdx.x] = sr;
    __syncthreads();
    for (int st = 128; st > 0; st >>= 1) {
        if (threadIdx.x < st) {
            redc[threadIdx.x] += redc[threadIdx.x + st];
            redr[threadIdx.x] += redr[threadIdx.x + st];
        }
        __syncthreads();
    }
    if (threadIdx.x == 0) {
        float cham_y = redc[0] / (float)(BATCH * MPTS);
        float cham_x = redr[0] / (float)(BATCH * NPTS);
        float lreg = 0.0f;
        for (int b = 0; b < BATCH; ++b)
            for (int i = 0; i < 3; ++i)
                for (int k = 0; k < 3; ++k) {
                    float acc = 0.0f;
                    for (int j = 0; j < 3; ++j)
                        acc += R_gt[b * 9 + j * 3 + i] * R[b * 9 + j * 3 + k];
                    acc -= (i == k) ? 1.0f : 0.0f;
                    lreg += acc * acc;
                }
        // diagonal(S, axis1=0, axis2=1): S_diag[k,i] = S[i,i,k]
        for (int i = 0; i < 3; ++i)
            for (int k = 0; k < 3; ++k) {
                float d = S[i * 9 + i * 3 + k] - S_gt[i * 9 + i * 3 + k];
                lreg += d * d;
            }
        for (int i = 0; i < BATCH * 3; ++i) {
            float d = t[i] - t_gt[i];
            lreg += d * d;
        }
        out[0] = lreg + cham_x + cham_y;
    }
}

extern "C" void kernel_launch(void* const* d_in, const int* in_sizes, int n_in,
                              void* d_out, int out_size, void* d_ws, size_t ws_size,
                              hipStream_t stream) {
    const float* x    = (const float*)d_in[0];
    const float* y    = (const float*)d_in[1];
    const float* R    = (const float*)d_in[2];
    const float* S    = (const float*)d_in[3];
    const float* t    = (const float*)d_in[4];
    const float* R_gt = (const float*)d_in[5];
    const float* S_gt = (const float*)d_in[6];
    const float* t_gt = (const float*)d_in[7];

    unsigned* colmin = (unsigned*)d_ws;          // BATCH*MPTS uints
    unsigned* rowmin = colmin + BATCH * MPTS;    // BATCH*NPTS uints

    hegn_init_kernel<<<(BATCH * (MPTS + NPTS) + 255) / 256, 256, 0, stream>>>(colmin);
    hegn_chamfer_kernel<<<BATCH * (NPTS / ROWS_PER_BLOCK) * MSPLIT, 256, 0, stream>>>(
        x, y, colmin, rowmin);
    hegn_finalize_kernel<<<1, 256, 0, stream>>>(
        colmin, rowmin, R, S, t, R_gt, S_gt, t_gt, (float*)d_out);
}